// InfoNCE_Loss_38036230373408
// MI455X (gfx1250) — compile-verified
//
#include <hip/hip_runtime.h>
#include <hip/hip_bf16.h>
#include <cmath>

typedef __attribute__((ext_vector_type(2))) float v2f;
typedef __attribute__((ext_vector_type(8))) float v8f;

#define N_ROWS   1026
#define N_COLSQ  104
#define DIMK     256
#define NBATCH   128
#define N_JTILE  65   // ceil(1026/16)
#define N_ITILE  7    // ceil(104/16)

// -------- main kernel: one (i-tile, batch) per block, 4 waves over j-tiles ----
__global__ __launch_bounds__(128)
void infonce_main_kernel(const float* __restrict__ H, const int* __restrict__ adj,
                         float* __restrict__ ws_neg, float* __restrict__ ws_s) {
    const int itile = blockIdx.x;          // 0..6
    const int b     = blockIdx.y;          // 0..127
    const int tid   = threadIdx.x;         // 0..127
    const int wave  = tid >> 5;
    const int lane  = tid & 31;
    const int n     = lane & 15;           // column / row-in-tile index
    const int half  = lane >> 4;           // 0 or 1

    __shared__ float lds_neg[16];
    if (tid < 16) lds_neg[tid] = 0.0f;
    __syncthreads();

    const float* Hb = H + (size_t)b * N_ROWS * DIMK;

    // ---- preload A fragments: Hq tile rows m = itile*16 + n ----
    // ISA 32-bit A 16x4 layout: lane (m + 16*half) holds K = 2*half, 2*half+1
    const float* arow = Hb + (size_t)(itile * 16 + n) * DIMK + 2 * half;
    v2f a[64];
#pragma unroll
    for (int f = 0; f < 64; ++f) {
        a[f] = *(const v2f*)(arow + 4 * f);
    }

    float accr[8];
#pragma unroll
    for (int r = 0; r < 8; ++r) accr[r] = 0.0f;

    // ---- j loop: this wave handles j-tiles wave, wave+4, ... ----
    for (int jt = wave; jt < N_JTILE; jt += 4) {
        const int jg = jt * 16 + n;                   // this lane's column j
        const int jc = (jg < N_ROWS) ? jg : (N_ROWS - 1);  // clamp address
        const float* brow = Hb + (size_t)jc * DIMK + 2 * half;

        v8f c0 = {}; v8f c1 = {};
#pragma unroll
        for (int f = 0; f < 64; f += 2) {
            v2f b0 = *(const v2f*)(brow + 4 * f);
            v2f b1 = *(const v2f*)(brow + 4 * (f + 1));
            c0 = __builtin_amdgcn_wmma_f32_16x16x4_f32(false, a[f],     false, b0,
                                                       (short)0, c0, false, false);
            c1 = __builtin_amdgcn_wmma_f32_16x16x4_f32(false, a[f + 1], false, b1,
                                                       (short)0, c1, false, false);
        }

        // mask[j,i] = (adj[j, i] == 0); rows handled by this lane: i = itile*16 + 8*half + r
        int mrow[8];
        const int ibase = itile * 16 + 8 * half;
        const bool valid = (jg < N_ROWS) && (ibase + 7 < N_COLSQ);
        if (valid) {
            const int* ap = adj + (size_t)jg * N_COLSQ + ibase;
            int4 m0 = *(const int4*)(ap);
            int4 m1 = *(const int4*)(ap + 4);
            mrow[0] = m0.x; mrow[1] = m0.y; mrow[2] = m0.z; mrow[3] = m0.w;
            mrow[4] = m1.x; mrow[5] = m1.y; mrow[6] = m1.z; mrow[7] = m1.w;
        } else {
#pragma unroll
            for (int r = 0; r < 8; ++r) mrow[r] = 1;   // nonzero adj -> mask 0
        }

#pragma unroll
        for (int r = 0; r < 8; ++r) {
            float cr = c0[r] + c1[r];
            float t  = __expf(2.0f * cr);
            accr[r] += (mrow[r] == 0) ? t : 0.0f;
        }
    }

    // ---- reduce lane partials into per-row sums (row = r + 8*half) ----
#pragma unroll
    for (int r = 0; r < 8; ++r) {
        atomicAdd(&lds_neg[r + 8 * half], accr[r]);
    }

    // ---- exact f32 positives: 16 threads, one Hq row each ----
    float s = 0.0f;
    if (tid < 16) {
        const float* row = Hb + (size_t)(itile * 16 + tid) * DIMK;
#pragma unroll 8
        for (int k = 0; k < DIMK; k += 4) {
            float4 v = *(const float4*)(row + k);
            s += v.x * v.x + v.y * v.y + v.z * v.z + v.w * v.w;
        }
    }
    __syncthreads();

    if (tid < 16) {
        const int iglob = itile * 16 + tid;
        if (iglob < N_COLSQ) {
            ws_neg[(size_t)iglob * NBATCH + b] = lds_neg[tid] + __expf(2.0f * s);
            ws_s  [(size_t)iglob * NBATCH + b] = s;
        }
    }
}

// -------- finalize: deterministic reduction over batches, log/finite/scale ----
__global__ __launch_bounds__(128)
void infonce_final_kernel(const float* __restrict__ ws_neg,
                          const float* __restrict__ ws_s,
                          float* __restrict__ out) {
    __shared__ float red[128];
    const int tid = threadIdx.x;
    float pc = 0.0f;
    if (tid < N_COLSQ) {
        float neg = 0.0f, ssum = 0.0f;
        for (int b = 0; b < NBATCH; ++b) {
            neg  += ws_neg[(size_t)tid * NBATCH + b];
            ssum += ws_s  [(size_t)tid * NBATCH + b];
        }
        float v = (float)NBATCH * __logf(neg) - 2.0f * ssum;
        pc = __builtin_isfinite(v) ? v : 0.0f;
    }
    red[tid] = pc;
    __syncthreads();
    for (int s = 64; s > 0; s >>= 1) {
        if (tid < s) red[tid] += red[tid + s];
        __syncthreads();
    }
    if (tid == 0) {
        out[0] = red[0] / ((float)N_ROWS * (float)NBATCH);
    }
}

extern "C" void kernel_launch(void* const* d_in, const int* in_sizes, int n_in,
                              void* d_out, int out_size, void* d_ws, size_t ws_size,
                              hipStream_t stream) {
    const float* H   = (const float*)d_in[0];   // (128, 1026, 256) f32
    const int*   adj = (const int*)d_in[1];     // (1026, 104) i32

    float* ws_neg = (float*)d_ws;                       // [104][128]
    float* ws_s   = ws_neg + (size_t)N_COLSQ * NBATCH;  // [104][128]

    dim3 grid(N_ITILE, NBATCH);
    infonce_main_kernel<<<grid, 128, 0, stream>>>(H, adj, ws_neg, ws_s);
    infonce_final_kernel<<<1, 128, 0, stream>>>(ws_neg, ws_s, (float*)d_out);
}